// SegmentedPolynomialProductJit_46256797778143
// MI455X (gfx1250) — compile-verified
//
#include <hip/hip_runtime.h>
#include <stdint.h>

// Problem constants (from reference): U=32, SA=4, SB=3, SC=4
#define UEXT   32
#define AROW   128          // SA*U floats per a-row
#define BROW   96           // SB*U floats per b-row
#define OROW   128          // SC*U floats per out-row
#define CHUNK  64           // edges staged per block iteration
#define TPB    256          // 8 waves of 32
#define BFLOATS (CHUNK * BROW)          // 6144 floats = 24KB per buffer
#define PACKETS (BFLOATS / 4)           // 16B packets per chunk = 1536
#define PASSES  (PACKETS / TPB)         // 6 async issues per thread

// ---- CDNA5 async global->LDS copy (gfx1250), tracked by ASYNCcnt ----
// b is streamed exactly once (384 MB) while the a-gather (51 MB) and the
// out-atomic (51 MB) footprints must stay resident in the 192 MB L2.
// TH_LOAD_NT streams b past L2 into LDS so it cannot evict the reusable set.
__device__ __forceinline__ void async_copy_b128_nt(uint32_t lds_addr, uint64_t gaddr) {
    asm volatile("global_load_async_to_lds_b128 %0, %1, off th:TH_LOAD_NT"
                 :: "v"(lds_addr), "v"(gaddr)
                 : "memory");
}

__device__ __forceinline__ void wait_async0() {
    asm volatile("s_wait_asynccnt 0" ::: "memory");
}

// Issue async copies for one 64-edge chunk of b into an LDS buffer.
__device__ __forceinline__ void stage_b_chunk(const float* __restrict__ b,
                                              int chunk, const float* ldsbuf,
                                              int tid, int E) {
    const long long gbaseF = (long long)chunk * CHUNK * BROW;     // float index
    const uint32_t lbase = (uint32_t)(uintptr_t)ldsbuf;           // LDS byte offset
    // floats remaining in this chunk (tail-safe; BROW is a multiple of 4)
    long long remEdges = (long long)E - (long long)chunk * CHUNK;
    long long limF = remEdges >= CHUNK ? (long long)BFLOATS : remEdges * BROW;
#pragma unroll
    for (int p = 0; p < PASSES; ++p) {
        int f = (p * TPB + tid) * 4;                              // float offset
        if (f < limF) {
            uint64_t g = (uint64_t)(uintptr_t)(b + gbaseF + f);
            async_copy_b128_nt(lbase + (uint32_t)(f * 4), g);
        }
    }
}

__global__ void __launch_bounds__(TPB)
seg_poly_zero(float4* __restrict__ out4, int n4) {
    const int stride = gridDim.x * TPB;
    for (int i = blockIdx.x * TPB + threadIdx.x; i < n4; i += stride)
        out4[i] = make_float4(0.f, 0.f, 0.f, 0.f);
}

__global__ void __launch_bounds__(TPB)
seg_poly_kernel(const float* __restrict__ a, const float* __restrict__ b,
                const int* __restrict__ idx_a, const int* __restrict__ idx_out,
                float* __restrict__ out, int E, int numChunks) {
    __shared__ float bbuf[2][BFLOATS];

    const int tid  = threadIdx.x;
    const int lane = tid & 31;     // u coordinate (U == wave32 width)
    const int wave = tid >> 5;     // 8 waves -> 8 edges each per chunk

    int chunk = blockIdx.x;
    const int stride = gridDim.x;
    if (chunk >= numChunks) return;

    // Prologue: fill buffer 0 with the first chunk.
    stage_b_chunk(b, chunk, &bbuf[0][0], tid, E);
    wait_async0();
    __syncthreads();

    int cur = 0;
    for (; chunk < numChunks; chunk += stride) {
        // Kick off the next chunk's async fill into the other buffer.
        const int nxt = chunk + stride;
        if (nxt < numChunks) stage_b_chunk(b, nxt, &bbuf[cur ^ 1][0], tid, E);

        // Compute this chunk from LDS. wave w handles edges w*8 .. w*8+7.
        const int base = chunk * CHUNK;
#pragma unroll
        for (int sub = 0; sub < CHUNK / 8; ++sub) {
            const int eL = wave * (CHUNK / 8) + sub;
            const int e  = base + eL;
            if (e < E) {
                const int ia = __builtin_amdgcn_readfirstlane(idx_a[e]);
                const int io = __builtin_amdgcn_readfirstlane(idx_out[e]);

                const float* ap = a + (size_t)ia * AROW + lane;   // L2-resident gather
                const float a0 = ap[0];
                const float a1 = ap[UEXT];
                const float a2 = ap[2 * UEXT];
                const float a3 = ap[3 * UEXT];

                const float* bp = &bbuf[cur][eL * BROW + lane];   // conflict-free LDS
                const float b0 = bp[0];
                const float b1 = bp[UEXT];
                const float b2 = bp[2 * UEXT];

                // coeff(i,j) = 0.1*(i+1) + 0.01*(j+1); k = (i+j) % 4
                float acc0 = 0.f, acc1 = 0.f, acc2 = 0.f, acc3 = 0.f;
                acc0 = fmaf(0.11f * a0, b0, acc0);   // i=0,j=0 -> k=0
                acc1 = fmaf(0.12f * a0, b1, acc1);   // i=0,j=1 -> k=1
                acc2 = fmaf(0.13f * a0, b2, acc2);   // i=0,j=2 -> k=2
                acc1 = fmaf(0.21f * a1, b0, acc1);   // i=1,j=0 -> k=1
                acc2 = fmaf(0.22f * a1, b1, acc2);   // i=1,j=1 -> k=2
                acc3 = fmaf(0.23f * a1, b2, acc3);   // i=1,j=2 -> k=3
                acc2 = fmaf(0.31f * a2, b0, acc2);   // i=2,j=0 -> k=2
                acc3 = fmaf(0.32f * a2, b1, acc3);   // i=2,j=1 -> k=3
                acc0 = fmaf(0.33f * a2, b2, acc0);   // i=2,j=2 -> k=0
                acc3 = fmaf(0.41f * a3, b0, acc3);   // i=3,j=0 -> k=3
                acc0 = fmaf(0.42f * a3, b1, acc0);   // i=3,j=1 -> k=0
                acc1 = fmaf(0.43f * a3, b2, acc1);   // i=3,j=2 -> k=1

                float* op = out + (size_t)io * OROW + lane;       // coalesced atomics
                atomicAdd(op,            acc0);
                atomicAdd(op + UEXT,     acc1);
                atomicAdd(op + 2 * UEXT, acc2);
                atomicAdd(op + 3 * UEXT, acc3);
            }
        }

        // Drain our async fills, then make the new buffer visible block-wide.
        wait_async0();
        __syncthreads();
        cur ^= 1;
    }
}

extern "C" void kernel_launch(void* const* d_in, const int* in_sizes, int n_in,
                              void* d_out, int out_size, void* d_ws, size_t ws_size,
                              hipStream_t stream) {
    (void)n_in; (void)d_ws; (void)ws_size;
    const float* a       = (const float*)d_in[0];
    const float* b       = (const float*)d_in[1];
    const int*   idx_a   = (const int*)d_in[2];
    const int*   idx_out = (const int*)d_in[3];
    float*       out     = (float*)d_out;

    const int E = in_sizes[2];                       // number of edges
    const int numChunks = (E + CHUNK - 1) / CHUNK;

    // Output is scatter-add accumulated: zero it first (harness poisons d_out).
    // out_size = M * 128, always divisible by 4.
    const int n4 = out_size / 4;
    int zBlocks = (n4 + TPB - 1) / TPB;
    if (zBlocks > 4096) zBlocks = 4096;
    if (zBlocks < 1) zBlocks = 1;
    seg_poly_zero<<<zBlocks, TPB, 0, stream>>>((float4*)out, n4);

    int blocks = numChunks < 2048 ? numChunks : 2048; // persistent, pipelined
    if (blocks < 1) blocks = 1;
    seg_poly_kernel<<<blocks, TPB, 0, stream>>>(a, b, idx_a, idx_out, out,
                                                E, numChunks);
}